// LongRangedAttention_59700045415020
// MI455X (gfx1250) — compile-verified
//
#include <hip/hip_runtime.h>
#include <hip/hip_bf16.h>

// LongRangedAttention — MI455X (gfx1250) version.
//
// The reference applies softmax over a size-1 axis, which is identically 1.0,
// so the attention masks are all-ones:
//   mask1 = mask2 = sum_c(1) = 256.0          (N,H,W)
//   v1    = spatial mean of l1_org            (N,C)
//   v2    = spatial mean of l2_org            (N,C)
// The whole 78M-element attention tensor is dead code. True cost: 3.2 MB of
// reads -> ~0.14us at 23.3 TB/s. The spatial mean is computed on the WMMA pipe
// (V_WMMA_F32_16X16X4_F32, exact f32) as x @ ones/HW.

typedef __attribute__((ext_vector_type(2))) float v2f;
typedef __attribute__((ext_vector_type(8))) float v8f;

#define N_  8
#define C_  256
#define HW_ 196

// out offsets (floats): mask1=0, v1=1568, mask2=3616, v2=5184; total 7232
#define OFF_V1    1568
#define OFF_MASK2 3616
#define OFF_V2    5184

// mask1/mask2: softmax over size-1 axis == 1.0, summed over C channels.
__global__ void lra_fill_masks(float* __restrict__ out) {
  int i = blockIdx.x * blockDim.x + threadIdx.x;
  if (i < N_ * HW_) {
    out[i]             = (float)C_;   // mask1
    out[OFF_MASK2 + i] = (float)C_;   // mask2
  }
}

// One wave32 per (tensor t, batch n, 16-channel block cb): 2*8*16 = 256 waves.
// Computes mean over HW=196 via 49 accumulating V_WMMA_F32_16X16X4_F32 ops
// with B filled with 1/HW (so every column of D is the row mean).
//
// A (16x4 f32) ISA layout: lane l holds row M = l&15;
//   VGPR0 = K = (l<16 ? 0 : 2), VGPR1 = K = (l<16 ? 1 : 3).
// D (16x16 f32) layout: VGPR v, lanes 0-15 -> (M=v, N=lane),
//   lanes 16-31 -> (M=v+8, N=lane-16). Column 0 lives in lanes 0 and 16.
__global__ __launch_bounds__(32) void lra_mean_wmma(const float* __restrict__ l1,
                                                    const float* __restrict__ l2,
                                                    float* __restrict__ out) {
  const int bx = blockIdx.x;        // 0..255
  const int t  = bx >> 7;           // tensor select
  const int n  = (bx >> 4) & 7;    // batch
  const int cb = bx & 15;           // channel block (16 channels)

  const int lane = threadIdx.x;     // wave32
  const int row  = lane & 15;       // M = channel within block
  const int hi   = lane >> 4;       // selects K pair {0,1} vs {2,3}

  const float* src = (t ? l2 : l1) +
                     (size_t)(n * C_ + cb * 16 + row) * HW_;
  // per-lane base: consecutive K pair, 8-byte aligned (HW_*4 and 2*hi*4 are
  // multiples of 8)
  const float2* p = (const float2*)(src + 2 * hi);

  v8f acc = {};
  v2f b;
  b.x = 1.0f / (float)HW_;
  b.y = 1.0f / (float)HW_;

#pragma unroll
  for (int k0 = 0; k0 < HW_; k0 += 4) {        // 49 iterations
    float2 d = p[k0 >> 1];                     // K = k0+2*hi, k0+2*hi+1
    v2f a;
    a.x = d.x;
    a.y = d.y;
    // 8 args: (neg_a, A, neg_b, B, c_mod, C, reuse_a, reuse_b)
    acc = __builtin_amdgcn_wmma_f32_16x16x4_f32(
        false, a, false, b, (short)0, acc, false, false);
  }

  // Column 0 of D: lane 0 holds rows 0..7 (acc[0..7]), lane 16 rows 8..15.
  if (row == 0) {
    float* dst = out + (t ? OFF_V2 : OFF_V1) + (size_t)n * C_ + cb * 16 + hi * 8;
#pragma unroll
    for (int v = 0; v < 8; ++v) dst[v] = acc[v];
  }
}

extern "C" void kernel_launch(void* const* d_in, const int* in_sizes, int n_in,
                              void* d_out, int out_size, void* d_ws, size_t ws_size,
                              hipStream_t stream) {
  (void)in_sizes; (void)n_in; (void)d_ws; (void)ws_size; (void)out_size;
  const float* l1 = (const float*)d_in[0];
  const float* l2 = (const float*)d_in[1];
  // d_in[2] (the 1x1 conv weight) provably does not affect the outputs.
  float* out = (float*)d_out;

  lra_fill_masks<<<(N_ * HW_ + 255) / 256, 256, 0, stream>>>(out);
  lra_mean_wmma<<<256, 32, 0, stream>>>(l1, l2, out);
}